// Align_30975304139293
// MI455X (gfx1250) — compile-verified
//
#include <hip/hip_runtime.h>

#define WPB 8              // waves per block
#define GPW 8              // frames per wave (one per lane in SVD phase)
#define NATOMS 128
#define NSEL 64
#define FF (NATOMS * 3)    // 384 floats per frame
#define XST 68             // padded row stride (floats) for transposed LDS tiles

typedef float v2f __attribute__((ext_vector_type(2)));
typedef float v8f __attribute__((ext_vector_type(8)));

struct alignas(16) WaveLds {
  float Xs_t[32 * XST];     // [m][k]: rows 0..11 frames 0..3 (coord-major), 12..15 zero,
                            //         rows 16..27 frames 4..7, 28..31 zero; cols k=0..63
  float Rf_t[16 * XST];     // [j][k]: rows 0..2 centered ref, row 3 = ones, rows 4..15 zero
  float Hb[GPW * 12];       // per frame: [0..8] 3x3 covariance H, [9..11] 64*centroid sums
};

__device__ __forceinline__ float wred32(float v) {
#pragma unroll
  for (int m = 16; m >= 1; m >>= 1) v += __shfl_xor(v, m, 32);
  return v;
}

// Symmetric Jacobi rotation on (p,q) with third index r; branchless, apq==0-safe.
__device__ __forceinline__ void jrotS(float& app, float& aqq, float& apq,
                                      float& arp, float& arq, float V[3][3],
                                      int p, int q) {
  float den = copysignf(fmaxf(fabsf(apq), 1e-30f), apq);
  float theta = 0.5f * (aqq - app) / den;
  float t = copysignf(1.0f, theta) / (fabsf(theta) + sqrtf(theta * theta + 1.0f));
  float c = rsqrtf(t * t + 1.0f);
  float s = t * c;
  float tau = t * apq;
  app -= tau;
  aqq += tau;
  apq = 0.0f;
  float rp = arp, rq = arq;
  arp = c * rp - s * rq;
  arq = s * rp + c * rq;
#pragma unroll
  for (int k = 0; k < 3; ++k) {
    float vkp = V[k][p], vkq = V[k][q];
    V[k][p] = c * vkp - s * vkq;
    V[k][q] = s * vkp + c * vkq;
  }
}

__global__ __launch_bounds__(WPB * 32) void align_kabsch_kernel(
    const float* __restrict__ traj, const float* __restrict__ ref_pos,
    const int* __restrict__ aidx, float* __restrict__ out, int nframes) {
  __shared__ WaveLds sh[WPB];
  const int lane = threadIdx.x & 31;
  const int wv = threadIdx.x >> 5;
  const int wave_base = (blockIdx.x * WPB + wv) * GPW;
  if (wave_base >= nframes) return;          // wave-uniform
  WaveLds& L = sh[wv];

  const float inv = 1.0f / (float)NSEL;
  const int i0 = aidx[lane];                 // this lane owns selected atoms k=lane, k=lane+32
  const int i1 = aidx[lane + 32];

  // ---- one-time zero fill of padding rows (cols 0..63) ----
  {
    float4 z4 = {0.0f, 0.0f, 0.0f, 0.0f};
    const int zr = 12 + (lane >> 3);         // rows 12..15
    const int zc = (lane & 7) * 8;           // cols 0..63, 8 per lane
    *(float4*)&L.Xs_t[zr * XST + zc] = z4;
    *(float4*)&L.Xs_t[zr * XST + zc + 4] = z4;
    *(float4*)&L.Xs_t[(zr + 16) * XST + zc] = z4;   // rows 28..31
    *(float4*)&L.Xs_t[(zr + 16) * XST + zc + 4] = z4;
#pragma unroll
    for (int rr = 0; rr < 3; ++rr) {         // Rf_t rows 4..15
      const int qr = 4 + rr * 4 + (lane >> 3);
      *(float4*)&L.Rf_t[qr * XST + zc] = z4;
      *(float4*)&L.Rf_t[qr * XST + zc + 4] = z4;
    }
  }

  // ---- centered reference + ones row (once per wave) ----
  float r0x = ref_pos[i0 * 3 + 0], r0y = ref_pos[i0 * 3 + 1], r0z = ref_pos[i0 * 3 + 2];
  float r1x = ref_pos[i1 * 3 + 0], r1y = ref_pos[i1 * 3 + 1], r1z = ref_pos[i1 * 3 + 2];
  float rcx = wred32(r0x + r1x) * inv;
  float rcy = wred32(r0y + r1y) * inv;
  float rcz = wred32(r0z + r1z) * inv;
  L.Rf_t[0 * XST + lane] = r0x - rcx;
  L.Rf_t[1 * XST + lane] = r0y - rcy;
  L.Rf_t[2 * XST + lane] = r0z - rcz;
  L.Rf_t[3 * XST + lane] = 1.0f;
  L.Rf_t[0 * XST + lane + 32] = r1x - rcx;
  L.Rf_t[1 * XST + lane + 32] = r1y - rcy;
  L.Rf_t[2 * XST + lane + 32] = r1z - rcz;
  L.Rf_t[3 * XST + lane + 32] = 1.0f;

  // ---- phase 1: gather selected atoms of 8 frames into transposed LDS tile ----
  // (raw coords: H = sum x(x)r' needs no centering since sum r' = 0; the centroid
  //  itself is produced by the WMMA ones row of B.)
#pragma unroll
  for (int g = 0; g < GPW; ++g) {
    int fb = wave_base + g;
    int fbe = (fb < nframes) ? fb : (nframes - 1);
    const float* fin = traj + (size_t)fbe * FF;
    __builtin_prefetch(fin + lane * 12, 0, 3);   // warm full frame for apply phase
    float p0x = fin[i0 * 3 + 0], p0y = fin[i0 * 3 + 1], p0z = fin[i0 * 3 + 2];
    float p1x = fin[i1 * 3 + 0], p1y = fin[i1 * 3 + 1], p1z = fin[i1 * 3 + 2];
    const int sb = (g < 4) ? (g * 3) : (16 + (g - 4) * 3);
    L.Xs_t[(sb + 0) * XST + lane] = p0x;
    L.Xs_t[(sb + 1) * XST + lane] = p0y;
    L.Xs_t[(sb + 2) * XST + lane] = p0z;
    L.Xs_t[(sb + 0) * XST + lane + 32] = p1x;
    L.Xs_t[(sb + 1) * XST + lane + 32] = p1y;
    L.Xs_t[(sb + 2) * XST + lane + 32] = p1z;
  }
  __asm__ volatile("" ::: "memory");

  // ---- phase 2: 8 covariances + 8 centroids via two f32 WMMA chains ----
  // A rows M = 3*f_local + i (12 used); B cols: j=0..2 centered ref, j=3 ones.
  // Transposed tiles make each fragment one contiguous b64 with immediate offsets.
#if defined(__AMDGCN__) && __has_builtin(__builtin_amdgcn_wmma_f32_16x16x4_f32)
  {
    const int hf = lane >> 4;
    const int lm = lane & 15;
    const float* xa0 = &L.Xs_t[lm * XST + 2 * hf];          // chain 0: frames 0..3
    const float* xa1 = &L.Xs_t[(16 + lm) * XST + 2 * hf];   // chain 1: frames 4..7
    const float* xb  = &L.Rf_t[lm * XST + 2 * hf];          // shared B fragment
    v8f acc0 = {}, acc1 = {};
#pragma unroll
    for (int k0 = 0; k0 < NSEL; k0 += 4) {
      v2f a0 = *(const v2f*)(xa0 + k0);
      v2f a1 = *(const v2f*)(xa1 + k0);
      v2f bv = *(const v2f*)(xb + k0);
      acc0 = __builtin_amdgcn_wmma_f32_16x16x4_f32(false, a0, false, bv, (short)0, acc0, false, false);
      acc1 = __builtin_amdgcn_wmma_f32_16x16x4_f32(false, a1, false, bv, (short)0, acc1, false, false);
    }
    // D: VGPR v, lane Lx -> (M = v + (Lx<16?0:8), N = Lx&15). Row r = 3f+i.
    // N=j<3: H[f][i][j]. N=3: sum_k x[k][i] = 64*centroid.
    const int rb = (lane < 16) ? 0 : 8;
    const int n = lane & 15;
#pragma unroll
    for (int v = 0; v < 8; ++v) {
      int r = rb + v;
      if (n < 4 && r < 12) {
        int f3 = r / 3;
        int i3 = r - f3 * 3;
        int slot = (n < 3) ? (i3 * 3 + n) : (9 + i3);
        L.Hb[f3 * 12 + slot] = acc0[v];             // frames 0..3
        L.Hb[4 * 12 + f3 * 12 + slot] = acc1[v];    // frames 4..7
      }
    }
  }
#else
  {
#pragma unroll
    for (int g = 0; g < GPW; ++g) {
      const int sb = (g < 4) ? (g * 3) : (16 + (g - 4) * 3);
      float a0[3] = { L.Xs_t[(sb + 0) * XST + lane], L.Xs_t[(sb + 1) * XST + lane], L.Xs_t[(sb + 2) * XST + lane] };
      float a1[3] = { L.Xs_t[(sb + 0) * XST + lane + 32], L.Xs_t[(sb + 1) * XST + lane + 32], L.Xs_t[(sb + 2) * XST + lane + 32] };
      float b0[3] = { L.Rf_t[0 * XST + lane], L.Rf_t[1 * XST + lane], L.Rf_t[2 * XST + lane] };
      float b1[3] = { L.Rf_t[0 * XST + lane + 32], L.Rf_t[1 * XST + lane + 32], L.Rf_t[2 * XST + lane + 32] };
#pragma unroll
      for (int i = 0; i < 3; ++i) {
#pragma unroll
        for (int j = 0; j < 3; ++j) {
          float pv = wred32(a0[i] * b0[j] + a1[i] * b1[j]);
          if (lane == 0) L.Hb[g * 12 + i * 3 + j] = pv;
        }
        float sv = wred32(a0[i] + a1[i]);
        if (lane == 0) L.Hb[g * 12 + 9 + i] = sv;
      }
    }
  }
#endif
  __asm__ volatile("" ::: "memory");

  // ---- phase 3: per-LANE Kabsch SVD (lane f handles frame f; lanes 8..31 duplicate) ----
  const int fl = lane & 7;
  float H[3][3];
#pragma unroll
  for (int i = 0; i < 3; ++i)
#pragma unroll
    for (int j = 0; j < 3; ++j) H[i][j] = L.Hb[fl * 12 + i * 3 + j];
  float xcf[3];
#pragma unroll
  for (int i = 0; i < 3; ++i) xcf[i] = L.Hb[fl * 12 + 9 + i] * inv;

  // K = H^T H, symmetric: 6 unique scalars
  float a00 = H[0][0]*H[0][0] + H[1][0]*H[1][0] + H[2][0]*H[2][0];
  float a11 = H[0][1]*H[0][1] + H[1][1]*H[1][1] + H[2][1]*H[2][1];
  float a22 = H[0][2]*H[0][2] + H[1][2]*H[1][2] + H[2][2]*H[2][2];
  float a01 = H[0][0]*H[0][1] + H[1][0]*H[1][1] + H[2][0]*H[2][1];
  float a02 = H[0][0]*H[0][2] + H[1][0]*H[1][2] + H[2][0]*H[2][2];
  float a12 = H[0][1]*H[0][2] + H[1][1]*H[1][2] + H[2][1]*H[2][2];
  float V[3][3] = {{1, 0, 0}, {0, 1, 0}, {0, 0, 1}};
#pragma unroll
  for (int sw = 0; sw < 4; ++sw) {
    jrotS(a00, a11, a01, a02, a12, V, 0, 1);   // r=2
    jrotS(a00, a22, a02, a01, a12, V, 0, 2);   // r=1
    jrotS(a11, a22, a12, a01, a02, V, 1, 2);   // r=0
  }

  float w0 = a00, w1 = a11, w2 = a22;
  if (w0 < w1) { float tw = w0; w0 = w1; w1 = tw;
#pragma unroll
    for (int i = 0; i < 3; ++i) { float tv = V[i][0]; V[i][0] = V[i][1]; V[i][1] = tv; } }
  if (w0 < w2) { float tw = w0; w0 = w2; w2 = tw;
#pragma unroll
    for (int i = 0; i < 3; ++i) { float tv = V[i][0]; V[i][0] = V[i][2]; V[i][2] = tv; } }
  if (w1 < w2) { float tw = w1; w1 = w2; w2 = tw;
#pragma unroll
    for (int i = 0; i < 3; ++i) { float tv = V[i][1]; V[i][1] = V[i][2]; V[i][2] = tv; } }

  float detV = V[0][0] * (V[1][1] * V[2][2] - V[1][2] * V[2][1])
             - V[0][1] * (V[1][0] * V[2][2] - V[1][2] * V[2][0])
             + V[0][2] * (V[1][0] * V[2][1] - V[1][1] * V[2][0]);
  if (detV < 0.0f) { V[0][2] = -V[0][2]; V[1][2] = -V[1][2]; V[2][2] = -V[2][2]; }

  float hv1[3], hv2[3];
#pragma unroll
  for (int i = 0; i < 3; ++i) hv1[i] = H[i][0] * V[0][0] + H[i][1] * V[1][0] + H[i][2] * V[2][0];
#pragma unroll
  for (int i = 0; i < 3; ++i) hv2[i] = H[i][0] * V[0][1] + H[i][1] * V[1][1] + H[i][2] * V[2][1];
  float n1 = rsqrtf(fmaxf(hv1[0] * hv1[0] + hv1[1] * hv1[1] + hv1[2] * hv1[2], 1e-30f));
  float u1[3] = { hv1[0] * n1, hv1[1] * n1, hv1[2] * n1 };
  float d12 = u1[0] * hv2[0] + u1[1] * hv2[1] + u1[2] * hv2[2];
#pragma unroll
  for (int i = 0; i < 3; ++i) hv2[i] -= d12 * u1[i];
  float n2 = rsqrtf(fmaxf(hv2[0] * hv2[0] + hv2[1] * hv2[1] + hv2[2] * hv2[2], 1e-30f));
  float u2[3] = { hv2[0] * n2, hv2[1] * n2, hv2[2] * n2 };
  float u3[3] = { u1[1] * u2[2] - u1[2] * u2[1],
                  u1[2] * u2[0] - u1[0] * u2[2],
                  u1[0] * u2[1] - u1[1] * u2[0] };
  float R[3][3];
#pragma unroll
  for (int i = 0; i < 3; ++i)
#pragma unroll
    for (int j = 0; j < 3; ++j)
      R[i][j] = u1[i] * V[j][0] + u2[i] * V[j][1] + u3[i] * V[j][2];
  // translation: q = p*R + t with t = -c*R  (folds centroid subtraction into FMAs)
  float tx = -(xcf[0] * R[0][0] + xcf[1] * R[1][0] + xcf[2] * R[2][0]);
  float ty = -(xcf[0] * R[0][1] + xcf[1] * R[1][1] + xcf[2] * R[2][1]);
  float tz = -(xcf[0] * R[0][2] + xcf[1] * R[1][2] + xcf[2] * R[2][2]);

  // ---- phase 4: apply rotation; R and t broadcast via readlane-style shuffles ----
#pragma unroll
  for (int g = 0; g < GPW; ++g) {
    int fb = wave_base + g;
    int fbe = (fb < nframes) ? fb : (nframes - 1);
    const float* fin = traj + (size_t)fbe * FF;
    float* fo = out + (size_t)fbe * FF;
    float Rg[3][3];
#pragma unroll
    for (int i = 0; i < 3; ++i)
#pragma unroll
      for (int j = 0; j < 3; ++j) Rg[i][j] = __shfl(R[i][j], g, 32);
    const float gx = __shfl(tx, g, 32);
    const float gy = __shfl(ty, g, 32);
    const float gz = __shfl(tz, g, 32);
    const bool ok = (fb < nframes);
#pragma unroll
    for (int t = 0; t < 4; ++t) {
      int a = lane + 32 * t;
      float px = fin[a * 3 + 0];
      float py = fin[a * 3 + 1];
      float pz = fin[a * 3 + 2];
      float qx = px * Rg[0][0] + py * Rg[1][0] + pz * Rg[2][0] + gx;
      float qy = px * Rg[0][1] + py * Rg[1][1] + pz * Rg[2][1] + gy;
      float qz = px * Rg[0][2] + py * Rg[1][2] + pz * Rg[2][2] + gz;
      if (ok) { fo[a * 3 + 0] = qx; fo[a * 3 + 1] = qy; fo[a * 3 + 2] = qz; }
    }
  }
}

extern "C" void kernel_launch(void* const* d_in, const int* in_sizes, int n_in,
                              void* d_out, int out_size, void* d_ws, size_t ws_size,
                              hipStream_t stream) {
  const float* traj = (const float*)d_in[0];
  const float* ref_pos = (const float*)d_in[1];
  const int* aidx = (const int*)d_in[2];
  float* out = (float*)d_out;
  int nframes = in_sizes[0] / FF;
  int frames_per_block = WPB * GPW;
  int blocks = (nframes + frames_per_block - 1) / frames_per_block;
  align_kabsch_kernel<<<blocks, WPB * 32, 0, stream>>>(traj, ref_pos, aidx, out, nframes);
}